// TTTLayerLinear_6356551598154
// MI455X (gfx1250) — compile-verified
//
#include <hip/hip_runtime.h>
#include <hip/hip_bf16.h>
#include <math.h>

// Dims (fixed by the reference)
#define B_   4
#define NM_  64
#define MB_  64
#define D_   1024
#define NH_  16
#define HD_  64
#define TOK_ (B_*NM_*MB_)          // 16384 tokens
#define TILE_FLOATS (MB_*HD_)      // 4096 floats per (b,h,n) tile

#define BASE_LR_INV 0.5413248546129181f

typedef __attribute__((ext_vector_type(2))) float v2f;
typedef __attribute__((ext_vector_type(8))) float v8f;

// ---------------------------------------------------------------------------
// V_WMMA_F32_16X16X4_F32 wrapper.  Fragment layout per CDNA5 ISA 7.12.2:
//   A (16x4):  lane L holds row m = L&15;  a.x = A[m][k0+2*(L>>4)], a.y = +1
//   B (4x16):  lane L holds col n = L&15;  b.x = B[k0+2*(L>>4)][n], b.y = +1
//   C/D:       vgpr r, lane L -> C[r + 8*(L>>4)][L&15]
// All B operands are staged k-contiguous ([n][k]) in LDS so both A and B
// fragments are single 64-bit LDS loads.
// ---------------------------------------------------------------------------
__device__ __forceinline__ v8f wmma4(v2f a, v2f b, v8f c) {
  return __builtin_amdgcn_wmma_f32_16x16x4_f32(false, a, false, b, (short)0, c,
                                               false, false);
}

// ===========================================================================
// Kernel 0: transpose the four 1024x1024 weight matrices into wt[n][k]
// grid (16,16,4), block 256; one 64x64 tile per block.
// ===========================================================================
__global__ __launch_bounds__(256) void ttt_transpose_kernel(
    const float* __restrict__ wq, const float* __restrict__ wk,
    const float* __restrict__ wv, const float* __restrict__ wo,
    float* __restrict__ wt)
{
  __shared__ float tls[64][68];
  const int mat = blockIdx.z;
  const float* src = (mat == 0) ? wq : (mat == 1) ? wk : (mat == 2) ? wv : wo;
  float* dst = wt + (size_t)mat * D_ * D_;
  const int k0 = blockIdx.y * 64;
  const int n0 = blockIdx.x * 64;
  const int t = threadIdx.x;
#pragma unroll
  for (int i = 0; i < 4; ++i) {
    const int idx = t + 256 * i;          // 0..1023
    const int r = idx >> 4, c4 = (idx & 15) * 4;
    *(float4*)&tls[r][c4] = *(const float4*)(src + (size_t)(k0 + r) * D_ + n0 + c4);
  }
  __syncthreads();
#pragma unroll
  for (int i = 0; i < 4; ++i) {
    const int idx = t + 256 * i;
    const int n = idx >> 4, r4 = (idx & 15) * 4;
    float4 v;
    v.x = tls[r4 + 0][n]; v.y = tls[r4 + 1][n];
    v.z = tls[r4 + 2][n]; v.w = tls[r4 + 3][n];
    *(float4*)(dst + (size_t)(n0 + n) * D_ + k0 + r4) = v;
  }
}

// ===========================================================================
// Kernel 1: QKV projection.  grid = (16 heads, 256 token-tiles, 3 matrices),
// block = 128 (4 waves).  B tile is loaded from the pre-transposed weights so
// fragments are contiguous b64 LDS reads.
// ===========================================================================
__global__ __launch_bounds__(128) void ttt_proj_kernel(
    const float* __restrict__ x, const float* __restrict__ wt,
    float* __restrict__ qbuf, float* __restrict__ kbuf,
    float* __restrict__ vbuf)
{
  __shared__ float As[64][36];    // 64 rows x 32 k  (pad 36: conflict-free, 16B aligned)
  __shared__ float Bst[64][36];   // 64 cols x 32 k  (transposed tile)

  const int h   = blockIdx.x;
  const int mt  = blockIdx.y;
  const int mat = blockIdx.z;
  const float* wtm = wt + (size_t)mat * D_ * D_;     // [n][k]
  float* ob = (mat == 0) ? qbuf : (mat == 1) ? kbuf : vbuf;

  const int b = mt >> 6, n = mt & 63;
  const float* xrow = x + (size_t)((b * NM_ + n) * MB_) * D_;
  const float* wrow = wtm + (size_t)(h * HD_) * D_;   // 64 transposed rows (= cols)

  const int t = threadIdx.x, lane = t & 31, wid = t >> 5;
  const int nl = lane & 15, hi = lane >> 4;
  const int arow = wid * 16 + nl;

  v8f acc[4];
#pragma unroll
  for (int ct = 0; ct < 4; ++ct) acc[ct] = (v8f)(0.0f);

  for (int kc = 0; kc < D_; kc += 32) {
#pragma unroll
    for (int i = 0; i < 4; ++i) {
      const int idx = t + 128 * i;          // 0..511
      const int r = idx >> 3, c4 = (idx & 7) * 4;
      *(float4*)&As[r][c4]  = *(const float4*)(xrow + (size_t)r * D_ + kc + c4);
      *(float4*)&Bst[r][c4] = *(const float4*)(wrow + (size_t)r * D_ + kc + c4);
    }
    if (kc + 32 < D_) {  // gfx1250 global_prefetch_b8 of next A chunk
      __builtin_prefetch(xrow + (size_t)(t >> 1) * D_ + kc + 32 + (t & 1) * 16, 0, 1);
    }
    __syncthreads();
#pragma unroll
    for (int k0 = 0; k0 < 32; k0 += 4) {
      v2f a = *(const v2f*)&As[arow][k0 + 2 * hi];
#pragma unroll
      for (int ct = 0; ct < 4; ++ct) {
        v2f bb = *(const v2f*)&Bst[ct * 16 + nl][k0 + 2 * hi];
        acc[ct] = wmma4(a, bb, acc[ct]);
      }
    }
    __syncthreads();
  }

  float* ob2 = ob + (size_t)(((b * NH_ + h) * NM_) + n) * TILE_FLOATS;
#pragma unroll
  for (int ct = 0; ct < 4; ++ct)
#pragma unroll
    for (int r = 0; r < 8; ++r) {
      const int row = wid * 16 + r + 8 * hi;
      const int col = ct * 16 + nl;
      ob2[row * HD_ + col] = acc[ct][r];
    }
}

// ===========================================================================
// Kernel 2: per-token learning rates  eta = clip(softplus(x@lr_w + lr_b + c))
// ===========================================================================
__global__ __launch_bounds__(256) void ttt_eta_kernel(
    const float* __restrict__ x, const float* __restrict__ lr_w,
    const float* __restrict__ lr_b, float* __restrict__ etabuf)
{
  const int t = threadIdx.x;
  const int h = t & 15;
  const size_t tok = (size_t)blockIdx.x * 16 + (t >> 4);
  const float* xr = x + tok * D_;
  float s = lr_b[h];
  for (int d = 0; d < D_; d += 4) {
    s += xr[d]     * lr_w[d * NH_ + h];
    s += xr[d + 1] * lr_w[(d + 1) * NH_ + h];
    s += xr[d + 2] * lr_w[(d + 2) * NH_ + h];
    s += xr[d + 3] * lr_w[(d + 3) * NH_ + h];
  }
  const float v = s + BASE_LR_INV;
  const float sp = (v > 20.0f) ? v : log1pf(expf(v));
  const float eta = fminf(fmaxf(sp, 0.0f), 1.0f);
  const int b = (int)(tok >> 12), n = (int)((tok >> 6) & 63), mb = (int)(tok & 63);
  etabuf[((size_t)(b * NH_ + h) * NM_ + n) * MB_ + mb] = eta;
}

// ===========================================================================
// Kernel 3: sequential TTT scan.  One 128-thread WG per (b,h) chain.
// Fast weight kept TRANSPOSED in LDS (w1t[j][k]) so GEMM1/3 B-fragments are
// b64 loads; z/edz tile kept transposed (est[j][m]) for the same reason in
// GEMM2, and so the 64 LN row-threads access LDS conflict-free.
// ===========================================================================
#define SCAN_PAD 68
#define SCAN_TILE (64 * SCAN_PAD)
#define SCAN_LDS_FLOATS (5 * SCAN_TILE + 5 * 64 + 16)
#define SCAN_LDS_BYTES (SCAN_LDS_FLOATS * sizeof(float))

__global__ __launch_bounds__(128) void ttt_scan_kernel(
    float* __restrict__ qbuf,            // read Q, write output in-place
    const float* __restrict__ kbuf, const float* __restrict__ vbuf,
    const float* __restrict__ etabuf, const unsigned char* __restrict__ mask,
    const float* __restrict__ W1, const float* __restrict__ b1,
    const float* __restrict__ norm_p)
{
  extern __shared__ float smem[];
  float (*w1t)[SCAN_PAD] = (float(*)[SCAN_PAD])(smem);                  // [j][k]
  float (*ks)[SCAN_PAD]  = (float(*)[SCAN_PAD])(smem + 1 * SCAN_TILE);  // [m][k]
  float (*qs)[SCAN_PAD]  = (float(*)[SCAN_PAD])(smem + 2 * SCAN_TILE);  // [m][k]
  float (*st)[SCAN_PAD]  = (float(*)[SCAN_PAD])(smem + 3 * SCAN_TILE);  // v-k, [m][j]
  float (*est)[SCAN_PAD] = (float(*)[SCAN_PAD])(smem + 4 * SCAN_TILE);  // z/edz/o, [j][m]
  float* b1s  = smem + 5 * SCAN_TILE;
  float* gs   = b1s + 64;
  float* bbs  = gs + 64;
  float* etas = bbs + 64;
  float* db1s = etas + 64;
  float* red  = db1s + 64;      // [0..3] wave partials, [4]=sclW, [5]=sclB

  const int bh = blockIdx.x;
  const int b = bh >> 4, h = bh & 15;
  const int t = threadIdx.x, lane = t & 31, wid = t >> 5;
  const int nl = lane & 15, hi = lane >> 4;
  const int arow = wid * 16 + nl;

  // init state (transposed) + per-head LN params
  for (int i = t; i < HD_ * HD_; i += 128)
    w1t[i & 63][i >> 6] = W1[h * 4096 + i];          // w1t[j][i] = W1[i][j]
  if (t < 64) {
    b1s[t] = b1[h * HD_ + t];
    gs[t]  = 1.0f + norm_p[h * 2 * HD_ + t];
    bbs[t] = norm_p[h * 2 * HD_ + HD_ + t];
  }
  __syncthreads();

  const size_t base0 = (size_t)bh * NM_ * TILE_FLOATS;

  for (int n = 0; n < NM_; ++n) {
    const size_t tb = base0 + (size_t)n * TILE_FLOATS;
#pragma unroll
    for (int i = 0; i < 8; ++i) {
      const int idx = t + 128 * i;          // 0..1023
      const int r = idx >> 4, c = (idx & 15) * 4;
      const float4 kv = *(const float4*)(kbuf + tb + r * HD_ + c);
      const float4 vv = *(const float4*)(vbuf + tb + r * HD_ + c);
      float4 sv;
      sv.x = vv.x - kv.x; sv.y = vv.y - kv.y;
      sv.z = vv.z - kv.z; sv.w = vv.w - kv.w;
      *(float4*)&ks[r][c] = kv;
      *(float4*)&st[r][c] = sv;
      *(float4*)&qs[r][c] = *(const float4*)(qbuf + tb + r * HD_ + c);
    }
    if (t < 64) etas[t] = etabuf[(size_t)bh * NM_ * MB_ + n * MB_ + t];
    const bool mm = mask[b * NM_ + n] != 0;
    __syncthreads();

    // ---- GEMM1: z = K @ W1 + b1 ----
    v8f acc[4];
#pragma unroll
    for (int ct = 0; ct < 4; ++ct) {
      const float bv = b1s[ct * 16 + nl];
      acc[ct] = (v8f){bv, bv, bv, bv, bv, bv, bv, bv};
    }
#pragma unroll
    for (int k0 = 0; k0 < HD_; k0 += 4) {
      v2f a = *(const v2f*)&ks[arow][k0 + 2 * hi];
#pragma unroll
      for (int ct = 0; ct < 4; ++ct) {
        v2f bb = *(const v2f*)&w1t[ct * 16 + nl][k0 + 2 * hi];
        acc[ct] = wmma4(a, bb, acc[ct]);
      }
    }
#pragma unroll
    for (int ct = 0; ct < 4; ++ct)
#pragma unroll
      for (int r = 0; r < 8; ++r)
        est[ct * 16 + nl][wid * 16 + r + 8 * hi] = acc[ct][r];   // z transposed
    __syncthreads();

    // ---- per-row LN forward + VJP -> edz (in place, transposed) ----
    if (t < 64) {
      float mu = 0.f, sq = 0.f;
      for (int j = 0; j < HD_; ++j) { const float z = est[j][t]; mu += z; sq += z * z; }
      mu *= (1.0f / HD_);
      const float var = sq * (1.0f / HD_) - mu * mu;
      const float rstd = rsqrtf(var + 1e-6f);
      float s1 = 0.f, s2 = 0.f;
      for (int j = 0; j < HD_; ++j) {
        const float xh = (est[j][t] - mu) * rstd;
        const float dy = gs[j] * ((xh * gs[j] + bbs[j]) - st[t][j]);
        s1 += dy; s2 += dy * xh;
      }
      s1 *= (1.0f / HD_); s2 *= (1.0f / HD_);
      const float e = etas[t];
      for (int j = 0; j < HD_; ++j) {
        const float xh = (est[j][t] - mu) * rstd;
        const float dy = gs[j] * ((xh * gs[j] + bbs[j]) - st[t][j]);
        est[j][t] = e * rstd * (dy - s1 - xh * s2);    // edz[m=t][j]
      }
    }
    __syncthreads();

    // ---- GEMM2: dW1 = K^T @ edz  (A[i][m] = K[m][i], B[m][j] = est[j][m]) ----
    v8f dacc[4];
#pragma unroll
    for (int ct = 0; ct < 4; ++ct) dacc[ct] = (v8f)(0.0f);
#pragma unroll
    for (int k0 = 0; k0 < HD_; k0 += 4) {
      v2f a;
      a.x = ks[k0 + 2 * hi][arow];
      a.y = ks[k0 + 2 * hi + 1][arow];
#pragma unroll
      for (int ct = 0; ct < 4; ++ct) {
        v2f bb = *(const v2f*)&est[ct * 16 + nl][k0 + 2 * hi];
        dacc[ct] = wmma4(a, bb, dacc[ct]);
      }
    }
    // Frobenius norm of dW1
    float ssq = 0.f;
#pragma unroll
    for (int ct = 0; ct < 4; ++ct)
#pragma unroll
      for (int r = 0; r < 8; ++r) { const float d = dacc[ct][r]; ssq += d * d; }
#pragma unroll
    for (int m = 16; m >= 1; m >>= 1) ssq += __shfl_xor(ssq, m, 32);
    if (lane == 0) red[wid] = ssq;
    // db1[j] = sum_m edz[m][j]  (est row j is contiguous)
    if (t < 64) { float sb = 0.f; for (int m = 0; m < MB_; ++m) sb += est[t][m]; db1s[t] = sb; }
    __syncthreads();
    if (t == 0) {
      const float nw = sqrtf(red[0] + red[1] + red[2] + red[3]);
      red[4] = mm ? ((nw > 1.0f) ? 1.0f / (nw + 1e-8f) : 1.0f) : 0.0f;
    }
    if (t == 1) {
      float sb = 0.f; for (int j = 0; j < HD_; ++j) sb += db1s[j] * db1s[j];
      const float nb = sqrtf(sb);
      red[5] = mm ? ((nb > 1.0f) ? 1.0f / (nb + 1e-8f) : 1.0f) : 0.0f;
    }
    __syncthreads();
    // ---- clipped update straight from the accumulators ----
    const float sclW = red[4];
#pragma unroll
    for (int ct = 0; ct < 4; ++ct)
#pragma unroll
      for (int r = 0; r < 8; ++r)
        w1t[ct * 16 + nl][wid * 16 + r + 8 * hi] -= sclW * dacc[ct][r];
    if (t < 64) b1s[t] -= red[5] * db1s[t];
    __syncthreads();

    // ---- GEMM3: o = Q @ W1' + b1' ----
#pragma unroll
    for (int ct = 0; ct < 4; ++ct) {
      const float bv = b1s[ct * 16 + nl];
      acc[ct] = (v8f){bv, bv, bv, bv, bv, bv, bv, bv};
    }
#pragma unroll
    for (int k0 = 0; k0 < HD_; k0 += 4) {
      v2f a = *(const v2f*)&qs[arow][k0 + 2 * hi];
#pragma unroll
      for (int ct = 0; ct < 4; ++ct) {
        v2f bb = *(const v2f*)&w1t[ct * 16 + nl][k0 + 2 * hi];
        acc[ct] = wmma4(a, bb, acc[ct]);
      }
    }
#pragma unroll
    for (int ct = 0; ct < 4; ++ct)
#pragma unroll
      for (int r = 0; r < 8; ++r)
        est[ct * 16 + nl][wid * 16 + r + 8 * hi] = acc[ct][r];   // o transposed
    __syncthreads();

    // ---- per-row LN + residual, write back over Q buffer ----
    if (t < 64) {
      float mu = 0.f, sq = 0.f;
      for (int j = 0; j < HD_; ++j) { const float z = est[j][t]; mu += z; sq += z * z; }
      mu *= (1.0f / HD_);
      const float var = sq * (1.0f / HD_) - mu * mu;
      const float rstd = rsqrtf(var + 1e-6f);
      for (int j = 0; j < HD_; ++j) {
        qbuf[tb + (size_t)t * HD_ + j] =
            (est[j][t] - mu) * rstd * gs[j] + bbs[j] + qs[t][j];
      }
    }
    __syncthreads();
  }
}

// ===========================================================================
// Kernel 4: post-LN + repack  Zh[b][h][n][mb][hd] -> Zn[token][d]
// ===========================================================================
__global__ __launch_bounds__(256) void ttt_postln_kernel(
    const float* __restrict__ zh, const float* __restrict__ post_scale,
    const float* __restrict__ post_bias, float* __restrict__ zn)
{
  const int wid = threadIdx.x >> 5, lane = threadIdx.x & 31;
  const size_t tok = (size_t)blockIdx.x * 8 + wid;
  const int b = (int)(tok >> 12), n = (int)((tok >> 6) & 63), mb = (int)(tok & 63);
  float vals[32];
  float s = 0.f, sq = 0.f;
#pragma unroll
  for (int j = 0; j < 32; ++j) {
    const int d = lane + 32 * j, hh = d >> 6, hd = d & 63;
    const float z =
        zh[(((size_t)(b * NH_ + hh) * NM_ + n) * MB_ + mb) * HD_ + hd];
    vals[j] = z; s += z; sq += z * z;
  }
#pragma unroll
  for (int m = 16; m >= 1; m >>= 1) { s += __shfl_xor(s, m, 32); sq += __shfl_xor(sq, m, 32); }
  const float mu = s * (1.0f / D_);
  const float var = sq * (1.0f / D_) - mu * mu;
  const float rstd = rsqrtf(var + 1e-6f);
#pragma unroll
  for (int j = 0; j < 32; ++j) {
    const int d = lane + 32 * j;
    zn[tok * D_ + d] = (vals[j] - mu) * rstd * post_scale[d] + post_bias[d];
  }
}

// ===========================================================================
// Kernel 5: output GEMM  out = x + Zn @ wo   (wo pre-transposed)
// ===========================================================================
__global__ __launch_bounds__(128) void ttt_out_kernel(
    const float* __restrict__ zn, const float* __restrict__ wot,
    const float* __restrict__ x, float* __restrict__ out)
{
  __shared__ float As[64][36];
  __shared__ float Bst[64][36];

  const int bx = blockIdx.x;
  const int mt = blockIdx.y;
  const float* arows = zn + (size_t)mt * 64 * D_;
  const float* wrow  = wot + (size_t)(bx * 64) * D_;   // transposed rows

  const int t = threadIdx.x, lane = t & 31, wid = t >> 5;
  const int nl = lane & 15, hi = lane >> 4;
  const int ar = wid * 16 + nl;

  v8f acc[4];
#pragma unroll
  for (int ct = 0; ct < 4; ++ct) acc[ct] = (v8f)(0.0f);

  for (int kc = 0; kc < D_; kc += 32) {
#pragma unroll
    for (int i = 0; i < 4; ++i) {
      const int idx = t + 128 * i;
      const int r = idx >> 3, c4 = (idx & 7) * 4;
      *(float4*)&As[r][c4]  = *(const float4*)(arows + (size_t)r * D_ + kc + c4);
      *(float4*)&Bst[r][c4] = *(const float4*)(wrow + (size_t)r * D_ + kc + c4);
    }
    if (kc + 32 < D_) {
      __builtin_prefetch(arows + (size_t)(t >> 1) * D_ + kc + 32 + (t & 1) * 16, 0, 1);
    }
    __syncthreads();
#pragma unroll
    for (int k0 = 0; k0 < 32; k0 += 4) {
      v2f a = *(const v2f*)&As[ar][k0 + 2 * hi];
#pragma unroll
      for (int ct = 0; ct < 4; ++ct) {
        v2f bb = *(const v2f*)&Bst[ct * 16 + nl][k0 + 2 * hi];
        acc[ct] = wmma4(a, bb, acc[ct]);
      }
    }
    __syncthreads();
  }

#pragma unroll
  for (int ct = 0; ct < 4; ++ct)
#pragma unroll
    for (int r = 0; r < 8; ++r) {
      const int row = mt * 64 + wid * 16 + r + 8 * hi;
      const int col = bx * 64 + ct * 16 + nl;
      const size_t o = (size_t)row * D_ + col;
      out[o] = acc[ct][r] + x[o];
    }
}

// ===========================================================================
extern "C" void kernel_launch(void* const* d_in, const int* in_sizes, int n_in,
                              void* d_out, int out_size, void* d_ws,
                              size_t ws_size, hipStream_t stream) {
  const float* x          = (const float*)d_in[0];
  const unsigned char* mk = (const unsigned char*)d_in[1];
  const float* wq         = (const float*)d_in[2];
  const float* wk         = (const float*)d_in[3];
  const float* wv         = (const float*)d_in[4];
  const float* wo         = (const float*)d_in[5];
  const float* lr_w       = (const float*)d_in[6];
  const float* lr_b       = (const float*)d_in[7];
  const float* W1         = (const float*)d_in[8];
  const float* b1         = (const float*)d_in[9];
  const float* norm_p     = (const float*)d_in[10];
  const float* post_scale = (const float*)d_in[11];
  const float* post_bias  = (const float*)d_in[12];
  float* out = (float*)d_out;

  const size_t TILES = (size_t)B_ * NH_ * NM_ * TILE_FLOATS;  // 16,777,216 floats
  float* qbuf   = (float*)d_ws;                   // Q, then scan output Zh
  float* kbuf   = qbuf + TILES;                   // K, then post-LN Zn
  float* vbuf   = kbuf + TILES;                   // V
  float* etabuf = vbuf + TILES;                   // [B,NH,NM,MB]
  float* wtbuf  = etabuf + (size_t)B_ * NH_ * NM_ * MB_;  // 4 transposed weights

  ttt_transpose_kernel<<<dim3(16, 16, 4), 256, 0, stream>>>(wq, wk, wv, wo,
                                                            wtbuf);
  ttt_proj_kernel<<<dim3(NH_, 256, 3), 128, 0, stream>>>(x, wtbuf, qbuf, kbuf,
                                                         vbuf);
  ttt_eta_kernel<<<TOK_ / 16, 256, 0, stream>>>(x, lr_w, lr_b, etabuf);
  ttt_scan_kernel<<<B_ * NH_, 128, SCAN_LDS_BYTES, stream>>>(
      qbuf, kbuf, vbuf, etabuf, mk, W1, b1, norm_p);
  ttt_postln_kernel<<<TOK_ / 8, 256, 0, stream>>>(qbuf, post_scale, post_bias,
                                                  kbuf);
  ttt_out_kernel<<<dim3(16, 256), 128, 0, stream>>>(
      kbuf + 0, wtbuf + (size_t)3 * D_ * D_, x, out);
}